// GEN_87239375716844
// MI455X (gfx1250) — compile-verified
//
#include <hip/hip_runtime.h>
#include <math.h>

#define N_NODES 50000
#define N_EDGES 640000
#define N_GRAPH 256
#define H       128
#define H2      256
#define LH      64
#define OUTC    10
#define L_LAYERS 5

typedef __attribute__((ext_vector_type(2))) float v2f;
typedef __attribute__((ext_vector_type(4))) float v4f;
typedef __attribute__((ext_vector_type(8))) float v8f;

// ---------------------------------------------------------------- utilities
__global__ void fill_f32(float* __restrict__ p, float v, int n) {
  int i = blockIdx.x * blockDim.x + threadIdx.x;
  if (i < n) p[i] = v;
}

__global__ void pack_xcat(const float* __restrict__ x, const float* __restrict__ cyc,
                          float* __restrict__ xcat, int n) {
  int i = blockIdx.x * blockDim.x + threadIdx.x;
  if (i >= n) return;
  int r = i / 40, c = i % 40;
  xcat[i] = (c < 32) ? x[r * 32 + c] : cyc[r * 8 + (c - 32)];
}

// ------------------------------------------------- fp32 WMMA GEMM: C = f(A)·Wᵀ + b
// A:[M,K] row-major, W:[Nc,K] row-major, M%16==0, Nc%64==0, K%4==0.
// One wave per block; each wave owns a 16x64 strip of C (4 independent WMMA
// accumulators -> 4-way ILP in the matrix pipe, A-fragment reused 4x).
// AFFINE: fuse per-K-channel y = relu(x*scale+shift) into the A load (BatchNorm+ReLU).
template<bool AFFINE>
__global__ void __launch_bounds__(32)
wmma_gemm_nt(const float* __restrict__ A, const float* __restrict__ W,
             const float* __restrict__ bias,
             const float* __restrict__ aScale, const float* __restrict__ aShift,
             float* __restrict__ C, int M, int Nc, int K)
{
  const int lane = threadIdx.x & 31;
  const int row0 = blockIdx.x << 4;
  const int col0 = blockIdx.y << 6;
  const int r    = lane & 15;
  const int koff = (lane >> 4) << 1;          // 0 for lanes 0-15, 2 for lanes 16-31

  const float* aRow = A + (size_t)(row0 + r) * K + koff;
  const float* w0   = W + (size_t)(col0 +  0 + r) * K + koff;
  const float* w1   = W + (size_t)(col0 + 16 + r) * K + koff;
  const float* w2   = W + (size_t)(col0 + 32 + r) * K + koff;
  const float* w3   = W + (size_t)(col0 + 48 + r) * K + koff;
  const float* sc   = AFFINE ? aScale + koff : nullptr;
  const float* sh   = AFFINE ? aShift + koff : nullptr;

  v8f acc0 = {0.f,0.f,0.f,0.f,0.f,0.f,0.f,0.f};
  v8f acc1 = acc0, acc2 = acc0, acc3 = acc0;

  #pragma unroll 2
  for (int k0 = 0; k0 < K; k0 += 4) {
    v2f a = *(const v2f*)(aRow + k0);
    if (AFFINE) {
      v2f s0 = *(const v2f*)(sc + k0);
      v2f s1 = *(const v2f*)(sh + k0);
      a.x = fmaxf(0.f, fmaf(a.x, s0.x, s1.x));
      a.y = fmaxf(0.f, fmaf(a.y, s0.y, s1.y));
    }
    v2f b0 = *(const v2f*)(w0 + k0);
    v2f b1 = *(const v2f*)(w1 + k0);
    v2f b2 = *(const v2f*)(w2 + k0);
    v2f b3 = *(const v2f*)(w3 + k0);
    acc0 = __builtin_amdgcn_wmma_f32_16x16x4_f32(false, a, false, b0, (short)0, acc0, false, false);
    acc1 = __builtin_amdgcn_wmma_f32_16x16x4_f32(false, a, false, b1, (short)0, acc1, false, false);
    acc2 = __builtin_amdgcn_wmma_f32_16x16x4_f32(false, a, false, b2, (short)0, acc2, false, false);
    acc3 = __builtin_amdgcn_wmma_f32_16x16x4_f32(false, a, false, b3, (short)0, acc3, false, false);
  }

  const int rbase = row0 + ((lane >> 4) << 3);
  float bv0 = bias[col0 +  0 + r];
  float bv1 = bias[col0 + 16 + r];
  float bv2 = bias[col0 + 32 + r];
  float bv3 = bias[col0 + 48 + r];
  #pragma unroll
  for (int v = 0; v < 8; ++v) {
    float* crow = C + (size_t)(rbase + v) * Nc + r;
    crow[col0 +  0] = acc0[v] + bv0;
    crow[col0 + 16] = acc1[v] + bv1;
    crow[col0 + 32] = acc2[v] + bv2;
    crow[col0 + 48] = acc3[v] + bv3;
  }
}

// ------------------------------------------------------------- edge passes
// msg = relu(h[src]+edge_emb[attr]) + eps  (> 0, recomputed on the fly).
// One thread handles 4 channels of one edge (b128 loads, 4 atomics).
__global__ void edge_max_pass(const float* __restrict__ h, const int* __restrict__ ei,
                              const int* __restrict__ eat, const float* __restrict__ emb,
                              int* __restrict__ mI)
{
  int idx = blockIdx.x * blockDim.x + threadIdx.x;      // E*32 threads
  if (idx >= N_EDGES * (H / 4)) return;
  int e = idx >> 5, c4 = (idx & 31) << 2;
  int src = ei[e], dst = ei[N_EDGES + e];
  v4f hv = *(const v4f*)(h + (size_t)src * H + c4);
  v4f ev = *(const v4f*)(emb + eat[e] * H + c4);
  int* mrow = mI + (size_t)dst * H + c4;
  #pragma unroll
  for (int j = 0; j < 4; ++j) {
    float msg = fmaxf(hv[j] + ev[j], 0.f) + 1e-7f;
    // msg > 0 and m initialized to 0.0f -> integer max == float max
    atomicMax(&mrow[j], __float_as_int(msg));
  }
}

__global__ void edge_sum_pass(const float* __restrict__ h, const int* __restrict__ ei,
                              const int* __restrict__ eat, const float* __restrict__ emb,
                              const float* __restrict__ m, float* __restrict__ s,
                              float* __restrict__ t)
{
  int idx = blockIdx.x * blockDim.x + threadIdx.x;
  if (idx >= N_EDGES * (H / 4)) return;
  int e = idx >> 5, c4 = (idx & 31) << 2;
  int src = ei[e], dst = ei[N_EDGES + e];
  size_t dj = (size_t)dst * H + c4;
  v4f hv = *(const v4f*)(h + (size_t)src * H + c4);
  v4f ev = *(const v4f*)(emb + eat[e] * H + c4);
  v4f mv = *(const v4f*)(m + dj);
  #pragma unroll
  for (int j = 0; j < 4; ++j) {
    float msg = fmaxf(hv[j] + ev[j], 0.f) + 1e-7f;
    float ex  = __expf(msg - mv[j]);
    atomicAdd(&s[dj + j], ex);          // sum of exp
    atomicAdd(&t[dj + j], ex * msg);    // sum of exp*msg -> agg = t/s
  }
}

// out = h + softmax-aggregated msg (empty segments contribute 0)
__global__ void combine(const float* __restrict__ h, const float* __restrict__ s,
                        const float* __restrict__ t, float* __restrict__ out, int n)
{
  int i = blockIdx.x * blockDim.x + threadIdx.x;
  if (i >= n) return;
  float sv = s[i];
  out[i] = h[i] + (sv > 0.f ? t[i] / sv : 0.f);
}

// -------------------------------------------------------------- batch norm
__global__ void bn_stats(const float* __restrict__ X, float* __restrict__ stats,
                         int Nrows, int C, int rowsPer)
{
  int c  = threadIdx.x;                      // blockDim.x == C
  int r0 = blockIdx.x * rowsPer;
  int r1 = min(r0 + rowsPer, Nrows);
  float s = 0.f, q = 0.f;
  for (int rr = r0; rr < r1; ++rr) {
    float v = X[(size_t)rr * C + c];
    s += v; q += v * v;
  }
  atomicAdd(&stats[c], s);
  atomicAdd(&stats[C + c], q);
}

__global__ void bn_final(const float* __restrict__ stats, const float* __restrict__ g,
                         const float* __restrict__ b, float* __restrict__ scale,
                         float* __restrict__ shift, int Nrows, int C)
{
  int c = threadIdx.x;
  if (c >= C) return;
  float invN = 1.f / (float)Nrows;
  float mu   = stats[c] * invN;
  float var  = stats[C + c] * invN - mu * mu;   // biased, matches x.var(0)
  float scv  = g[c] * rsqrtf(var + 1e-5f);
  scale[c] = scv;
  shift[c] = b[c] - mu * scv;
}

__global__ void bn_relu_apply(const float* __restrict__ X, const float* __restrict__ scale,
                              const float* __restrict__ shift, float* __restrict__ Y, int n)
{
  int i = blockIdx.x * blockDim.x + threadIdx.x;
  if (i >= n) return;
  int c = i & (H - 1);
  Y[i] = fmaxf(0.f, fmaf(X[i], scale[c], shift[c]));
}

// ------------------------------------------------------------------- pooling
__global__ void pool_sum(const float* __restrict__ h, const int* __restrict__ batch,
                         float* __restrict__ p)
{
  int i = blockIdx.x * blockDim.x + threadIdx.x;
  if (i >= N_NODES * H) return;
  int n = i >> 7, c = i & (H - 1);
  atomicAdd(&p[(size_t)batch[n] * H + c], h[i]);
}

__global__ void add_bias2(const float* __restrict__ a, const float* __restrict__ b,
                          float* __restrict__ o)
{
  int i = threadIdx.x;
  o[i] = a[i] + b[i];
}

// ------------------------------------------- single-step LSTM (h0=c0=0) + head
__global__ void lstm_head(const float* __restrict__ gates, const float* __restrict__ lin_w,
                          const float* __restrict__ lin_b, float* __restrict__ out)
{
  __shared__ float hh[LH];
  int g = blockIdx.x, l = threadIdx.x;
  const float* row = gates + (size_t)g * (4 * LH);
  float ig = row[l];                 // input gate
  float gg = row[2 * LH + l];        // cell candidate (forget gate unused: c0=0)
  float og = row[3 * LH + l];        // output gate
  float cc = (1.f / (1.f + __expf(-ig))) * tanhf(gg);
  hh[l]    = (1.f / (1.f + __expf(-og))) * tanhf(cc);
  __syncthreads();
  if (l < OUTC) {
    float acc = lin_b[l];
    #pragma unroll
    for (int k = 0; k < LH; ++k) acc += hh[k] * lin_w[l * LH + k];
    out[g * OUTC + l] = acc;
  }
}

// =========================================================================
extern "C" void kernel_launch(void* const* d_in, const int* in_sizes, int n_in,
                              void* d_out, int out_size, void* d_ws, size_t ws_size,
                              hipStream_t stream)
{
  const float* x     = (const float*)d_in[0];
  const float* cyc   = (const float*)d_in[1];
  const int*   ei    = (const int*)  d_in[2];
  const int*   eat   = (const int*)  d_in[3];
  const int*   batch = (const int*)  d_in[4];
  const float* fc_w  = (const float*)d_in[5];
  const float* fc_b  = (const float*)d_in[6];
  const float* emb   = (const float*)d_in[7];
  const float* cw1   = (const float*)d_in[8];
  const float* cb1   = (const float*)d_in[9];
  const float* cbn_g = (const float*)d_in[10];
  const float* cbn_b = (const float*)d_in[11];
  const float* cw2   = (const float*)d_in[12];
  const float* cb2   = (const float*)d_in[13];
  const float* bn_g  = (const float*)d_in[14];
  const float* bn_b  = (const float*)d_in[15];
  const float* wih   = (const float*)d_in[16];
  /* d_in[17] = lstm_whh unused: h0 == 0 */
  const float* bih   = (const float*)d_in[18];
  const float* bhh   = (const float*)d_in[19];
  const float* lin_w = (const float*)d_in[20];
  const float* lin_b = (const float*)d_in[21];
  float* out = (float*)d_out;
  (void)in_sizes; (void)n_in; (void)out_size; (void)ws_size;

  // ---- workspace carve-out (fp32) ----
  float* Wp = (float*)d_ws;
  size_t o = 0;
  float* hA   = Wp + o; o += (size_t)N_NODES * H;    // node features (persists)
  float* mB   = Wp + o; o += (size_t)N_NODES * H;    // seg-max, then combine output
  float* sC   = Wp + o; o += (size_t)N_NODES * H;    // sum(exp), then conv output
  float* tD   = Wp + o; o += (size_t)N_NODES * H;    // sum(exp*msg)
  float* mid  = Wp + o; o += (size_t)N_NODES * H2;   // MLP hidden; aliases xcat
  float* stats= Wp + o; o += 512;
  float* scl1 = Wp + o; o += 256;
  float* shf1 = Wp + o; o += 256;
  float* scl2 = Wp + o; o += 128;
  float* shf2 = Wp + o; o += 128;
  float* pooled = Wp + o; o += (size_t)N_GRAPH * H;
  float* gates  = Wp + o; o += (size_t)N_GRAPH * 4 * LH;
  float* bsum   = Wp + o; o += 4 * LH;

  const int TB = 256;
  const int NH = N_NODES * H;
  const int nhB = (NH + TB - 1) / TB;
  const int EH4 = N_EDGES * (H / 4);                 // edge threads (4 ch each)
  const int ehB = (EH4 + TB - 1) / TB;

  // ---- input FC: h = [x | cyc] @ fc_wᵀ + fc_b ----
  pack_xcat<<<(N_NODES * 40 + TB - 1) / TB, TB, 0, stream>>>(x, cyc, mid, N_NODES * 40);
  wmma_gemm_nt<false><<<dim3(N_NODES / 16, H / 64), 32, 0, stream>>>(
      mid, fc_w, fc_b, nullptr, nullptr, hA, N_NODES, H, 40);

  // ---- 5 GENConv layers ----
  for (int i = 0; i < L_LAYERS; ++i) {
    fill_f32<<<nhB, TB, 0, stream>>>(mB, 0.f, NH);
    fill_f32<<<nhB, TB, 0, stream>>>(sC, 0.f, NH);
    fill_f32<<<nhB, TB, 0, stream>>>(tD, 0.f, NH);
    edge_max_pass<<<ehB, TB, 0, stream>>>(hA, ei, eat, emb, (int*)mB);
    edge_sum_pass<<<ehB, TB, 0, stream>>>(hA, ei, eat, emb, mB, sC, tD);
    combine<<<nhB, TB, 0, stream>>>(hA, sC, tD, mB, NH);   // mB = agg + h

    // MLP Linear(H,2H)
    wmma_gemm_nt<false><<<dim3(N_NODES / 16, H2 / 64), 32, 0, stream>>>(
        mB, cw1 + (size_t)i * H2 * H, cb1 + i * H2, nullptr, nullptr,
        mid, N_NODES, H2, H);
    // inner BN stats + params; BN+ReLU fused into second GEMM's A load
    fill_f32<<<2, TB, 0, stream>>>(stats, 0.f, 512);
    bn_stats<<<(N_NODES + 127) / 128, H2, 0, stream>>>(mid, stats, N_NODES, H2, 128);
    bn_final<<<1, H2, 0, stream>>>(stats, cbn_g + i * H2, cbn_b + i * H2, scl1, shf1,
                                   N_NODES, H2);
    wmma_gemm_nt<true><<<dim3(N_NODES / 16, H / 64), 32, 0, stream>>>(
        mid, cw2 + (size_t)i * H * H2, cb2 + i * H, scl1, shf1,
        sC, N_NODES, H, H2);
    // outer BN + ReLU -> hA for next layer
    fill_f32<<<1, TB, 0, stream>>>(stats, 0.f, 256);
    bn_stats<<<(N_NODES + 127) / 128, H, 0, stream>>>(sC, stats, N_NODES, H, 128);
    bn_final<<<1, H, 0, stream>>>(stats, bn_g + i * H, bn_b + i * H, scl2, shf2,
                                  N_NODES, H);
    bn_relu_apply<<<nhB, TB, 0, stream>>>(sC, scl2, shf2, hA, NH);
  }

  // ---- global add pool, LSTM step, head ----
  fill_f32<<<(N_GRAPH * H + TB - 1) / TB, TB, 0, stream>>>(pooled, 0.f, N_GRAPH * H);
  pool_sum<<<nhB, TB, 0, stream>>>(hA, batch, pooled);
  add_bias2<<<1, 4 * LH, 0, stream>>>(bih, bhh, bsum);
  wmma_gemm_nt<false><<<dim3(N_GRAPH / 16, (4 * LH) / 64), 32, 0, stream>>>(
      pooled, wih, bsum, nullptr, nullptr, gates, N_GRAPH, 4 * LH, H);
  lstm_head<<<N_GRAPH, LH, 0, stream>>>(gates, lin_w, lin_b, out);
}